// VNUpdate_38001870635465
// MI455X (gfx1250) — compile-verified
//
#include <hip/hip_runtime.h>

#define N_NODES  100000
#define N_GRAPHS 64
#define DIM      1024
#define BN_EPS   1e-5f

typedef __attribute__((ext_vector_type(2))) float v2f;
typedef __attribute__((ext_vector_type(8))) float v8f;

// ---------------------------------------------------------------------------
// K0: xbuf = vn_h   (64 x 1024)
// ---------------------------------------------------------------------------
__global__ void k_init_x(const float* __restrict__ vn_h, float* __restrict__ xbuf) {
    int i = blockIdx.x * blockDim.x + threadIdx.x;
    xbuf[i] = vn_h[i];
}

// ---------------------------------------------------------------------------
// K1: xbuf[g] += sum over nodes of graph g  (batch is sorted -> register
//     accumulate within a 256-node chunk, atomic flush only on run change)
//     grid: (ceil(N/256), DIM/256), block 256. Thread owns one dim column.
// ---------------------------------------------------------------------------
__global__ void k_segsum(const float* __restrict__ h, const int* __restrict__ batch,
                         float* __restrict__ xbuf) {
    __shared__ int sb[256];
    const int tid = threadIdx.x;
    const int n0  = blockIdx.x * 256;
    const int d   = blockIdx.y * 256 + tid;
    {
        int idx = n0 + tid;
        sb[tid] = (idx < N_NODES) ? batch[idx] : -1;
    }
    __syncthreads();

    const int cnt = (N_NODES - n0) < 256 ? (N_NODES - n0) : 256;
    float acc = 0.f;
    int cur = sb[0];
    for (int i = 0; i < cnt; ++i) {
        int b = sb[i];
        if (b != cur) {
            atomicAdd(&xbuf[cur * DIM + d], acc);
            acc = 0.f;
            cur = b;
        }
        acc += h[(size_t)(n0 + i) * DIM + d];
    }
    atomicAdd(&xbuf[cur * DIM + d], acc);
}

// ---------------------------------------------------------------------------
// K2: y = x @ W1^T  via V_WMMA_F32_16X16X4_F32.
//     y[m][n] = sum_k x[m][k] * W1[n][k]  -> B[k][n] = W1[n][k], so the
//     B-operand per lane reads row n of W1 (same index math as A).
//     One wave -> one 16x16 tile; 256 tiles total; 32 blocks x 8 waves.
// ---------------------------------------------------------------------------
__global__ void k_gemm1(const float* __restrict__ x, const float* __restrict__ W1,
                        float* __restrict__ y) {
    const int lane  = threadIdx.x & 31;
    const int wave  = threadIdx.x >> 5;
    const int tile  = blockIdx.x * 8 + wave;   // 0..255
    const int tileM = tile >> 6;               // 0..3   (M = 64 / 16)
    const int tileN = tile & 63;               // 0..63  (N = 1024 / 16)

    const int r  = lane & 15;                  // A: M row / B: N col
    const int kh = (lane >> 4) << 1;           // lane half -> K {0,1} or {2,3}

    const float* A = x  + (size_t)(tileM * 16 + r) * DIM;
    const float* B = W1 + (size_t)(tileN * 16 + r) * DIM;

    v8f acc = {};
    for (int k = 0; k < DIM; k += 4) {
        v2f a = *(const v2f*)(A + k + kh);
        v2f b = *(const v2f*)(B + k + kh);
        acc = __builtin_amdgcn_wmma_f32_16x16x4_f32(false, a, false, b,
                                                    (short)0, acc, false, false);
    }

    // D layout: VGPR i, lanes 0-15 -> M=i, lanes 16-31 -> M=i+8; N = lane&15
    const int mb  = tileM * 16 + ((lane >> 4) << 3);
    const int col = tileN * 16 + (lane & 15);
    for (int i = 0; i < 8; ++i)
        y[(size_t)(mb + i) * DIM + col] = acc[i];
}

// ---------------------------------------------------------------------------
// K3: per-column BN coefficients over the 64 rows of y (biased variance):
//     scale[j] = gamma[j] * rsqrt(var + eps);  shift[j] = beta[j] - mean*scale
// ---------------------------------------------------------------------------
__global__ void k_bnstats(const float* __restrict__ y, const float* __restrict__ gamma,
                          const float* __restrict__ beta, float* __restrict__ scale,
                          float* __restrict__ shift) {
    int j = blockIdx.x * blockDim.x + threadIdx.x;   // column 0..1023
    float s = 0.f, sq = 0.f;
    for (int r = 0; r < N_GRAPHS; ++r) {
        float v = y[(size_t)r * DIM + j];
        s  += v;
        sq += v * v;
    }
    const float inv = 1.f / (float)N_GRAPHS;
    float mean = s * inv;
    float var  = sq * inv - mean * mean;
    float rstd = rsqrtf(var + BN_EPS);
    float sc   = rstd * gamma[j];
    scale[j] = sc;
    shift[j] = beta[j] - mean * sc;
}

// ---------------------------------------------------------------------------
// K4: vn_new = relu(BN(y)) @ W2^T  -- BN+ReLU fused into the A-operand load.
// ---------------------------------------------------------------------------
__global__ void k_gemm2(const float* __restrict__ y, const float* __restrict__ scale,
                        const float* __restrict__ shift, const float* __restrict__ W2,
                        float* __restrict__ vn_new) {
    const int lane  = threadIdx.x & 31;
    const int wave  = threadIdx.x >> 5;
    const int tile  = blockIdx.x * 8 + wave;
    const int tileM = tile >> 6;
    const int tileN = tile & 63;

    const int r  = lane & 15;
    const int kh = (lane >> 4) << 1;

    const float* A = y  + (size_t)(tileM * 16 + r) * DIM;
    const float* B = W2 + (size_t)(tileN * 16 + r) * DIM;

    v8f acc = {};
    for (int k = 0; k < DIM; k += 4) {
        const int kk = k + kh;
        v2f a;
        a[0] = fmaxf(fmaf(A[kk],     scale[kk],     shift[kk]),     0.f);
        a[1] = fmaxf(fmaf(A[kk + 1], scale[kk + 1], shift[kk + 1]), 0.f);
        v2f b = *(const v2f*)(B + kk);
        acc = __builtin_amdgcn_wmma_f32_16x16x4_f32(false, a, false, b,
                                                    (short)0, acc, false, false);
    }

    const int mb  = tileM * 16 + ((lane >> 4) << 3);
    const int col = tileN * 16 + (lane & 15);
    for (int i = 0; i < 8; ++i)
        vn_new[(size_t)(mb + i) * DIM + col] = acc[i];
}

// ---------------------------------------------------------------------------
// K5: h_new = h + vn_new[batch]   -- the bandwidth kernel (~820 MB).
//     One block per node, 256 threads x float4 = one 4 KB row; vn rows
//     (256 KB total) stay resident in cache.
// ---------------------------------------------------------------------------
__global__ void k_broadcast(const float* __restrict__ h, const int* __restrict__ batch,
                            const float* __restrict__ vn, float* __restrict__ out) {
    const int n = blockIdx.x;
    const int t = threadIdx.x;                 // 0..255 float4 slots
    const int b = batch[n];
    const float4* hv = (const float4*)(h  + (size_t)n * DIM);
    const float4* vv = (const float4*)(vn + (size_t)b * DIM);
    float4*       ov = (float4*)(out + (size_t)n * DIM);
    float4 a = hv[t];
    float4 c = vv[t];
    ov[t] = make_float4(a.x + c.x, a.y + c.y, a.z + c.z, a.w + c.w);
}

// ---------------------------------------------------------------------------
extern "C" void kernel_launch(void* const* d_in, const int* in_sizes, int n_in,
                              void* d_out, int out_size, void* d_ws, size_t ws_size,
                              hipStream_t stream) {
    const float* h     = (const float*)d_in[0];
    const int*   batch = (const int*)  d_in[1];
    const float* vn_h  = (const float*)d_in[2];
    const float* W1    = (const float*)d_in[3];
    const float* gamma = (const float*)d_in[4];
    const float* beta  = (const float*)d_in[5];
    const float* W2    = (const float*)d_in[6];

    float* out    = (float*)d_out;
    float* out_vn = out + (size_t)N_NODES * DIM;   // second tuple element

    float* ws    = (float*)d_ws;
    float* xbuf  = ws;                      // 64*1024
    float* ybuf  = ws + 65536;              // 64*1024
    float* scale = ws + 131072;             // 1024
    float* shift = ws + 132096;             // 1024

    k_init_x<<<(N_GRAPHS * DIM) / 256, 256, 0, stream>>>(vn_h, xbuf);

    dim3 gseg((N_NODES + 255) / 256, DIM / 256);
    k_segsum<<<gseg, 256, 0, stream>>>(h, batch, xbuf);

    k_gemm1<<<32, 256, 0, stream>>>(xbuf, W1, ybuf);

    k_bnstats<<<DIM / 256, 256, 0, stream>>>(ybuf, gamma, beta, scale, shift);

    k_gemm2<<<32, 256, 0, stream>>>(ybuf, scale, shift, W2, out_vn);

    k_broadcast<<<N_NODES, 256, 0, stream>>>(h, batch, out_vn, out);
}